// RoutingBlock_30640296689903
// MI455X (gfx1250) — compile-verified
//
#include <hip/hip_runtime.h>

// Reference shapes (fixed by the harness):
//   inputs:    (B=64, H=32, W=32, C=512) fp32
//   routing_x: (B=64, ROUTES=4)          fp32
//   output:    (B=64, H=32, W=32, RW=128) fp32, RW = C/ROUTES
//
// Pure gather: out[b,h,w,:] = inputs[b,h,w, route(b)*128 : +128]
// HBM-bound: 32 MiB in + 32 MiB out => ~2.7 us at 23.3 TB/s.

typedef __attribute__((ext_vector_type(4))) float f4;

namespace {
constexpr int kRoutes       = 4;
constexpr int kRouteWidth   = 128;                 // floats per route slice
constexpr int kC            = kRoutes * kRouteWidth; // 512
constexpr int kRowsPerBatch = 32 * 32;             // H*W = 1024 rows per batch
constexpr int kRowsPerBlock = 8;                   // 8 waves * (1 row / wave32)
constexpr int kTotalRows    = 64 * kRowsPerBatch;  // B*H*W = 65536
}  // namespace

__global__ __launch_bounds__(256) void routing_gather_kernel(
    const f4* __restrict__ in,        // inputs viewed as float4: row stride = C/4 = 128
    const float* __restrict__ rx,     // routing logits, B x 4
    f4* __restrict__ out)             // output viewed as float4: row stride = RW/4 = 32
{
    // ---- Uniform (scalar) routing decision for this block ---------------
    // kRowsPerBlock (8) divides kRowsPerBatch (1024), so every row this
    // block touches belongs to the same batch. blockIdx-derived values are
    // wave-uniform -> compiler can keep this on the scalar path (s_load +
    // uniform compares), no LDS / no barrier needed.
    const int row0 = blockIdx.x * kRowsPerBlock;
    const int b    = row0 / kRowsPerBatch;

    const float v0 = rx[b * kRoutes + 0];
    const float v1 = rx[b * kRoutes + 1];
    const float v2 = rx[b * kRoutes + 2];
    const float v3 = rx[b * kRoutes + 3];

    // jnp.argmax semantics: first occurrence of the max wins (strict '>').
    int route = 0;
    float best = v0;
    if (v1 > best) { best = v1; route = 1; }
    if (v2 > best) { best = v2; route = 2; }
    if (v3 > best) { best = v3; route = 3; }

    // ---- One wave32 per 128-float output row ----------------------------
    // Lane l moves float4 #l of the row: the wave issues one fully
    // coalesced 512B b128 load and one 512B b128 store.
    const int wave = threadIdx.x >> 5;   // 0..7
    const int lane = threadIdx.x & 31;   // 0..31

    const size_t row = (size_t)row0 + (size_t)wave;   // global pixel row

    const f4* src = in  + row * (kC / 4)          // start of this pixel's C=512 floats
                        + (size_t)route * (kRouteWidth / 4)  // selected 128-float slice
                        + lane;
    f4* dst       = out + row * (kRouteWidth / 4) + lane;

    // Streamed exactly once each way -> non-temporal policy (TH=NT) keeps
    // the 64 MiB stream from displacing useful L2 contents.
    const f4 v = __builtin_nontemporal_load(src);
    __builtin_nontemporal_store(v, dst);
}

extern "C" void kernel_launch(void* const* d_in, const int* in_sizes, int n_in,
                              void* d_out, int out_size, void* d_ws, size_t ws_size,
                              hipStream_t stream) {
    (void)in_sizes; (void)n_in; (void)out_size; (void)d_ws; (void)ws_size;

    const f4*    in  = (const f4*)d_in[0];     // (64,32,32,512) fp32
    const float* rx  = (const float*)d_in[1];  // (64,4) fp32
    f4*          out = (f4*)d_out;             // (64,32,32,128) fp32

    const int blocks = kTotalRows / kRowsPerBlock;   // 65536 / 8 = 8192
    routing_gather_kernel<<<blocks, 256, 0, stream>>>(in, rx, out);
}